// GCNModel_23003844838151
// MI455X (gfx1250) — compile-verified
//
#include <hip/hip_runtime.h>
#include <hip/hip_bf16.h>

#define N_NODES 50000
#define N_EDGES 800000
#define D 128

typedef float v2f __attribute__((ext_vector_type(2)));
typedef float v8f __attribute__((ext_vector_type(8)));

// ---------------------------------------------------------------------------
// Zero-fill a float buffer (float4 granularity; n must be multiple of 4)
// ---------------------------------------------------------------------------
__global__ void zero_f4_kernel(float4* __restrict__ p, int n4) {
    int i = blockIdx.x * blockDim.x + threadIdx.x;
    if (i < n4) p[i] = make_float4(0.f, 0.f, 0.f, 0.f);
}

// ---------------------------------------------------------------------------
// Degree accumulation: one thread per edge, float atomics
// ---------------------------------------------------------------------------
__global__ void degree_kernel(const int* __restrict__ src,
                              const int* __restrict__ dst,
                              float* __restrict__ out_deg,
                              float* __restrict__ in_deg,
                              int n_edges) {
    int e = blockIdx.x * blockDim.x + threadIdx.x;
    if (e < n_edges) {
        atomicAdd(&out_deg[src[e]], 1.0f);
        atomicAdd(&in_deg[dst[e]], 1.0f);
    }
}

// deg -> rsqrt(max(deg, 1)) in place (covers both arrays as one span)
__global__ void deg_finalize_kernel(float* __restrict__ deg, int n) {
    int i = blockIdx.x * blockDim.x + threadIdx.x;
    if (i < n) deg[i] = rsqrtf(fmaxf(deg[i], 1.0f));
}

// ---------------------------------------------------------------------------
// Fused edge scatter: one wave32 per edge. Each lane moves 4 contiguous
// floats: coalesced 512B float4 gather of H[src], scaled in-register by
// inv_out[src] (src-side normalization fused — no staging buffer), then 4
// global_atomic_add_f32 into M[dst]. The whole working set (h 25.6MB,
// M 25.6MB, edges 6.4MB) is L2-resident, so atomics resolve at L2 rate.
// ---------------------------------------------------------------------------
__global__ void edge_scatter_kernel(const float* __restrict__ H,
                                    const float* __restrict__ inv_out,
                                    const int* __restrict__ src,
                                    const int* __restrict__ dst,
                                    float* __restrict__ M, int n_edges) {
    int gid  = blockIdx.x * blockDim.x + threadIdx.x;
    int edge = gid >> 5;
    int lane = gid & 31;
    if (edge >= n_edges) return;
    int s = src[edge];
    int d = dst[edge];
    float sc = inv_out[s];
    float4 v = ((const float4*)(H + (size_t)s * D))[lane];
    float* mrow = M + (size_t)d * D + lane * 4;
    atomicAdd(mrow + 0, v.x * sc);
    atomicAdd(mrow + 1, v.y * sc);
    atomicAdd(mrow + 2, v.z * sc);
    atomicAdd(mrow + 3, v.w * sc);
}

// ---------------------------------------------------------------------------
// OUT = relu( (M * inv_in[row]) @ W + b )  via V_WMMA_F32_16X16X4_F32.
// Block = 256 threads = 8 waves; wave w computes the 16x16 tile at
// (rowtile = blockIdx.x, coltile = w). K = 128 in steps of 4 (32 WMMAs).
//
// A 16x4 f32 layout (ISA 7.12.2): lanes 0-15 -> M=lane, VGPR0=K0, VGPR1=K1;
//                                 lanes 16-31 -> M=lane-16, VGPR0=K2, VGPR1=K3.
// C/D 16x16 f32: VGPR r: lanes 0-15 -> (M=r, N=lane); lanes 16-31 -> (M=r+8).
// ---------------------------------------------------------------------------
__global__ __launch_bounds__(256)
void gemm_bias_relu_kernel(const float* __restrict__ M,
                           const float* __restrict__ inv_in,
                           const float* __restrict__ W,
                           const float* __restrict__ b,
                           float* __restrict__ OUT) {
    const int wave = threadIdx.x >> 5;   // 0..7 -> column tile
    const int lane = threadIdx.x & 31;
    const int half = lane >> 4;          // 0: K pair {k,k+1}; 1: {k+2,k+3}
    const int l    = lane & 15;
    const int rt   = blockIdx.x * 16;
    const int ct   = wave * 16;

    const int   arow = rt + l;
    const float s    = inv_in[arow];                 // fused dst-normalization
    const float* Ar  = M + (size_t)arow * D;

    v8f c = {0.f, 0.f, 0.f, 0.f, 0.f, 0.f, 0.f, 0.f};

    #pragma unroll
    for (int k0 = 0; k0 < D; k0 += 4) {
        const int ka = k0 + half * 2;
        v2f a, bb;
        a.x  = Ar[ka]     * s;
        a.y  = Ar[ka + 1] * s;
        bb.x = W[(size_t)ka       * D + ct + l];
        bb.y = W[(size_t)(ka + 1) * D + ct + l];
        c = __builtin_amdgcn_wmma_f32_16x16x4_f32(
                /*neg_a=*/false, a, /*neg_b=*/false, bb,
                /*c_mod=*/(short)0, c, /*reuse_a=*/false, /*reuse_b=*/false);
    }

    const int   col  = ct + l;
    const float bias = b[col];
    #pragma unroll
    for (int r = 0; r < 8; ++r) {
        const int row = rt + r + half * 8;
        OUT[(size_t)row * D + col] = fmaxf(c[r] + bias, 0.0f);
    }
}

// ---------------------------------------------------------------------------
// Launcher: 3 GraphConv layers. Workspace layout (floats):
//   [0, 50000)            : out_deg  -> inv_out
//   [50000, 100000)       : in_deg   -> inv_in
//   [100000, 100000+6.4M) : M  (aggregated messages)
// d_out doubles as the inter-layer H buffer (GEMM never reads d_out).
// ---------------------------------------------------------------------------
extern "C" void kernel_launch(void* const* d_in, const int* in_sizes, int n_in,
                              void* d_out, int out_size, void* d_ws, size_t ws_size,
                              hipStream_t stream) {
    const float* h   = (const float*)d_in[0];
    const int*   src = (const int*)  d_in[1];
    const int*   dst = (const int*)  d_in[2];
    const float* Wm[3] = { (const float*)d_in[3], (const float*)d_in[5], (const float*)d_in[7] };
    const float* bv[3] = { (const float*)d_in[4], (const float*)d_in[6], (const float*)d_in[8] };
    float* out = (float*)d_out;

    float* ws      = (float*)d_ws;
    float* inv_out = ws;                       // 50000
    float* inv_in  = ws + N_NODES;             // 50000
    float* Mbuf    = ws + 2 * N_NODES;         // 50000*128

    const int TPB = 256;

    // ---- degrees (graph is identical across layers: compute once) ----
    {
        int n4 = (2 * N_NODES) / 4;
        zero_f4_kernel<<<(n4 + TPB - 1) / TPB, TPB, 0, stream>>>((float4*)inv_out, n4);
        degree_kernel<<<(N_EDGES + TPB - 1) / TPB, TPB, 0, stream>>>(src, dst, inv_out, inv_in, N_EDGES);
        deg_finalize_kernel<<<(2 * N_NODES + TPB - 1) / TPB, TPB, 0, stream>>>(inv_out, 2 * N_NODES);
    }

    const int nfeat4    = N_NODES * (D / 4);                 // float4 count of a feature matrix
    const int zero_grid = (nfeat4 + TPB - 1) / TPB;
    const int scat_grid = ((N_EDGES * 32) + TPB - 1) / TPB;  // one wave32 per edge
    const int gemm_grid = N_NODES / 16;                      // 3125 row tiles

    const float* Hin = h;
    for (int layer = 0; layer < 3; ++layer) {
        zero_f4_kernel<<<zero_grid, TPB, 0, stream>>>((float4*)Mbuf, nfeat4);
        edge_scatter_kernel<<<scat_grid, TPB, 0, stream>>>(Hin, inv_out, src, dst, Mbuf, N_EDGES);
        gemm_bias_relu_kernel<<<gemm_grid, TPB, 0, stream>>>(Mbuf, inv_in, Wm[layer], bv[layer], out);
        Hin = out;   // d_out is the H buffer for the next layer
    }
}